// NMS_8985071583743
// MI455X (gfx1250) — compile-verified
//
#include <hip/hip_runtime.h>
#include <hip/hip_bf16.h>

#define B       8
#define N       25200
#define NC      80
#define ROW     85
#define K       2048
#define CAND    4096
#define MAXDET  300
#define CONF_T  0.3f
#define IOU_T   0.6f

typedef __attribute__((address_space(3))) char lds_char;

// ---- CDNA5 async global->LDS copies (ASYNCcnt path), inline asm (portable) ----
__device__ __forceinline__ void async_ld_b128(void* lds_dst, const void* gsrc) {
    unsigned loff = (unsigned)(size_t)(lds_char*)lds_dst;
    unsigned long long ga = (unsigned long long)(size_t)gsrc;
    asm volatile("global_load_async_to_lds_b128 %0, %1, off"
                 :: "v"(loff), "v"(ga) : "memory");
}
__device__ __forceinline__ void async_ld_b32(void* lds_dst, const void* gsrc) {
    unsigned loff = (unsigned)(size_t)(lds_char*)lds_dst;
    unsigned long long ga = (unsigned long long)(size_t)gsrc;
    asm volatile("global_load_async_to_lds_b32 %0, %1, off"
                 :: "v"(loff), "v"(ga) : "memory");
}
__device__ __forceinline__ void wait_asynccnt0() {
    asm volatile("s_wait_asynccnt 0" ::: "memory");
}

// ================= Kernel 1: per-row score / class (wave32 per row) =========
__global__ __launch_bounds__(256)
void score_kernel(const float* __restrict__ x,
                  float* __restrict__ scores, int* __restrict__ clsid) {
    const int lane = threadIdx.x & 31;
    const int wid  = threadIdx.x >> 5;
    const long long row = (long long)blockIdx.x * (blockDim.x >> 5) + wid;
    if (row >= (long long)B * N) return;

    const float* p = x + (size_t)row * ROW;
    const float obj = p[4];

    float best = -1.0f; int bestc = 0;
    for (int c = lane; c < NC; c += 32) {
        float v = p[5 + c] * obj;
        if (v > best) { best = v; bestc = c; }
    }
    // wave32 max+argmax reduction (lowest class index on ties)
    for (int off = 16; off > 0; off >>= 1) {
        float ob = __shfl_down(best, off, 32);
        int   oc = __shfl_down(bestc, off, 32);
        if (ob > best || (ob == best && oc < bestc)) { best = ob; bestc = oc; }
    }
    if (lane == 0) {
        const bool valid = (obj > CONF_T) && (best > CONF_T);
        scores[row] = valid ? best : -1.0f;
        clsid[row]  = bestc;
    }
}

// ============ Kernel 2: per-batch top-K via histogram + LDS bitonic =========
__global__ __launch_bounds__(1024)
void topk_kernel(const float* __restrict__ x,
                 const float* __restrict__ scores, const int* __restrict__ clsid,
                 float4* __restrict__ selbox, float* __restrict__ selscore,
                 int* __restrict__ selcls) {
    const int b   = blockIdx.x;
    const int tid = threadIdx.x;
    const int nt  = blockDim.x;

    __shared__ int   hist[256];
    __shared__ float cs[CAND];
    __shared__ int   ci[CAND];
    __shared__ int   cnt;
    __shared__ int   thrBin;

    for (int i = tid; i < 256; i += nt) hist[i] = 0;
    if (tid == 0) { cnt = 0; thrBin = 0; }
    __syncthreads();

    const float* sb = scores + (size_t)b * N;
    for (int i = tid; i < N; i += nt) {
        float s = sb[i];
        if (s > 0.0f) {
            int bin = (int)(s * 256.0f); if (bin > 255) bin = 255;
            atomicAdd(&hist[bin], 1);
        }
    }
    __syncthreads();

    if (tid == 0) {
        int acc = 0, tb = 0;
        for (int v = 255; v >= 0; --v) {
            acc += hist[v];
            if (acc >= K) { tb = v; break; }
        }
        thrBin = tb;
    }
    __syncthreads();
    const int T = thrBin;

    for (int i = tid; i < N; i += nt) {
        float s = sb[i];
        if (s > 0.0f) {
            int bin = (int)(s * 256.0f); if (bin > 255) bin = 255;
            if (bin >= T) {
                int p = atomicAdd(&cnt, 1);
                if (p < CAND) { cs[p] = s; ci[p] = i; }
            }
        }
    }
    __syncthreads();
    const int total = (cnt < CAND) ? cnt : CAND;
    for (int i = tid; i < CAND; i += nt)
        if (i >= total) { cs[i] = -1.0e30f; ci[i] = -1; }
    __syncthreads();

    // bitonic sort, descending by score
    for (int ks = 2; ks <= CAND; ks <<= 1) {
        for (int j = ks >> 1; j > 0; j >>= 1) {
            for (int i = tid; i < CAND; i += nt) {
                int l = i ^ j;
                if (l > i) {
                    float si = cs[i], sl = cs[l];
                    bool sw = ((i & ks) == 0) ? (si < sl) : (si > sl);
                    if (sw) {
                        cs[i] = sl; cs[l] = si;
                        int t = ci[i]; ci[i] = ci[l]; ci[l] = t;
                    }
                }
            }
            __syncthreads();
        }
    }

    // gather top-K rows, xywh -> xyxy
    for (int i = tid; i < K; i += nt) {
        float s = cs[i]; int idx = ci[i];
        float4 bx; int c = 0;
        if (idx >= 0 && s > 0.0f) {
            const float* r = x + ((size_t)b * N + idx) * ROW;
            float cx = r[0], cy = r[1], hw = r[2] * 0.5f, hh = r[3] * 0.5f;
            bx.x = cx - hw; bx.y = cy - hh; bx.z = cx + hw; bx.w = cy + hh;
            c = clsid[(size_t)b * N + idx];
        } else {
            bx.x = bx.y = bx.z = bx.w = 0.0f; s = -1.0f;
        }
        selbox[(size_t)b * K + i]   = bx;
        selscore[(size_t)b * K + i] = s;
        selcls[(size_t)b * K + i]   = c;
    }
}

// ========== Kernel 3: per-batch greedy NMS (async-staged LDS) + output ======
__global__ __launch_bounds__(1024)
void nms_kernel(const float4* __restrict__ selbox, const float* __restrict__ selscore,
                const int* __restrict__ selcls, float* __restrict__ out) {
    const int b   = blockIdx.x;
    const int tid = threadIdx.x;
    const int nt  = blockDim.x;

    __shared__ float4 lbox[K];            // 32 KB
    __shared__ float  lsc[K];             //  8 KB
    __shared__ int    lcls[K];            //  8 KB
    __shared__ unsigned char keep[K];     //  2 KB
    __shared__ int nv_sh;

    const float4* gb = selbox   + (size_t)b * K;
    const float*  gs = selscore + (size_t)b * K;
    const int*    gc = selcls   + (size_t)b * K;

    // async global -> LDS staging (ASYNCcnt)
    for (int i = tid; i < K; i += nt) async_ld_b128(&lbox[i], &gb[i]);
    for (int i = tid; i < K; i += nt) async_ld_b32(&lsc[i], &gs[i]);
    for (int i = tid; i < K; i += nt) async_ld_b32(&lcls[i], &gc[i]);
    wait_asynccnt0();
    __syncthreads();

    if (tid == 0) nv_sh = 0;
    for (int i = tid; i < K; i += nt) keep[i] = 1;
    __syncthreads();
    {
        int c = 0;
        for (int i = tid; i < K; i += nt) if (lsc[i] > 0.0f) c++;
        if (c) atomicAdd(&nv_sh, c);
    }
    __syncthreads();
    const int nv = nv_sh;          // sorted descending -> valid is a prefix

    // greedy suppression: serial in i, parallel in j
    for (int i = 0; i < nv; ++i) {
        if (keep[i]) {
            const float4 a = lbox[i];
            const float areaA = (a.z - a.x) * (a.w - a.y);
            const int ca = lcls[i];
            for (int j = i + 1 + tid; j < nv; j += nt) {
                if (keep[j] && lcls[j] == ca) {
                    const float4 bb = lbox[j];
                    float lx = fmaxf(a.x, bb.x), ly = fmaxf(a.y, bb.y);
                    float rx = fminf(a.z, bb.z), ry = fminf(a.w, bb.w);
                    float inter = fmaxf(rx - lx, 0.0f) * fmaxf(ry - ly, 0.0f);
                    float areaB = (bb.z - bb.x) * (bb.w - bb.y);
                    float iou = inter / (areaA + areaB - inter + 1e-9f);
                    if (iou > IOU_T) keep[j] = 0;
                }
            }
        }
        __syncthreads();
    }

    // emit detections: dets (B,300,6) then flags (B,300)
    if (tid == 0) {
        float* dets  = out + (size_t)b * MAXDET * 6;
        float* flags = out + (size_t)B * MAXDET * 6 + (size_t)b * MAXDET;
        int oc = 0;
        for (int i = 0; i < nv && oc < MAXDET; ++i) {
            if (keep[i]) {
                const float4 bx = lbox[i];
                dets[oc * 6 + 0] = bx.x; dets[oc * 6 + 1] = bx.y;
                dets[oc * 6 + 2] = bx.z; dets[oc * 6 + 3] = bx.w;
                dets[oc * 6 + 4] = lsc[i];
                dets[oc * 6 + 5] = (float)lcls[i];
                flags[oc] = 1.0f;
                ++oc;
            }
        }
        for (; oc < MAXDET; ++oc) {
            dets[oc * 6 + 0] = 0.0f; dets[oc * 6 + 1] = 0.0f;
            dets[oc * 6 + 2] = 0.0f; dets[oc * 6 + 3] = 0.0f;
            dets[oc * 6 + 4] = -1.0f; dets[oc * 6 + 5] = 0.0f;
            flags[oc] = 0.0f;
        }
    }
}

extern "C" void kernel_launch(void* const* d_in, const int* in_sizes, int n_in,
                              void* d_out, int out_size, void* d_ws, size_t ws_size,
                              hipStream_t stream) {
    const float* x = (const float*)d_in[0];
    float* out = (float*)d_out;
    char* ws = (char*)d_ws;

    float*  scores   = (float*)ws;                               // B*N
    int*    clsid    = (int*)(ws + (size_t)B * N * 4);           // B*N
    float4* selbox   = (float4*)(ws + (size_t)B * N * 8);        // B*K (16B aligned)
    float*  selscore = (float*)((char*)selbox + (size_t)B * K * 16);
    int*    selcls   = (int*)((char*)selscore + (size_t)B * K * 4);

    const int rows_per_block = 256 / 32;                         // 8 waves/block
    dim3 g1((B * N + rows_per_block - 1) / rows_per_block);
    score_kernel<<<g1, 256, 0, stream>>>(x, scores, clsid);
    topk_kernel<<<B, 1024, 0, stream>>>(x, scores, clsid, selbox, selscore, selcls);
    nms_kernel<<<B, 1024, 0, stream>>>(selbox, selscore, selcls, out);
}